// FieldAwareInteractionLayer_11974368821309
// MI455X (gfx1250) — compile-verified
//
#include <hip/hip_runtime.h>

#define FIELDS 39
#define EMB 16
#define NPAIR 741                      // 39*38/2
#define ROWF (FIELDS * EMB)            // 624 floats per table row
#define ROWB (ROWF * 4)                // 2496 bytes per table row
#define CH_PER_ROW (ROWB / 16)         // 156 16-byte chunks per row
#define TOT_CH (FIELDS * CH_PER_ROW)   // 6084 chunks per sample
#define OUT_F4 ((NPAIR * EMB) / 4)     // 2964 float4 outputs per sample
#define BLOCK 256

// Compile-time (i,j) pair table matching np.triu_indices(39, k=1) order.
struct PairTab { unsigned short v[NPAIR]; };
static constexpr PairTab make_pairs() {
  PairTab t{};
  int p = 0;
  for (int i = 0; i < FIELDS; ++i)
    for (int j = i + 1; j < FIELDS; ++j)
      t.v[p++] = (unsigned short)((i << 8) | j);
  return t;
}
__constant__ PairTab c_pairs = make_pairs();

// Types matching the async-to-LDS builtin's parameters:
//   param0: int4 in global (AS1) address space, non-const
//   param1: int4 in LDS (AS3) address space
typedef int v4i __attribute__((ext_vector_type(4)));
typedef v4i __attribute__((address_space(1))) *gv4p;
typedef v4i __attribute__((address_space(3))) *lv4p;

typedef float v4f __attribute__((ext_vector_type(4)));

__global__ __launch_bounds__(BLOCK) void ffm_interact_kernel(
    const int* __restrict__ X, const float* __restrict__ table,
    float* __restrict__ out) {
  __shared__ float        s_rows[FIELDS * ROWF];   // 97,344 B: 39 gathered rows
  __shared__ unsigned int s_rowoff[FIELDS];        // byte offset of each row

  const int b   = blockIdx.x;
  const int tid = threadIdx.x;

  // Phase 1: row byte-offsets from the 39 indices of this sample.
  if (tid < FIELDS) {
    s_rowoff[tid] = (unsigned int)X[b * FIELDS + tid] * (unsigned int)ROWB;
  }
  __syncthreads();

  // Phase 2: gather 39 rows (6084 x 16B) into LDS via the async copy engine
  // (global_load_async_to_lds_b128, tracked by ASYNCcnt). Table reads keep
  // default temporal hint so the table stays resident in the 192MB L2.
  const char* tbase = (const char*)table;
#if __has_builtin(__builtin_amdgcn_global_load_async_to_lds_b128)
  for (int c = tid; c < TOT_CH; c += BLOCK) {
    const int f = c / CH_PER_ROW;
    const int r = c - f * CH_PER_ROW;
    char* g = (char*)tbase + (size_t)s_rowoff[f] + (size_t)((unsigned)r << 4);
    char* l = (char*)s_rows + ((size_t)c << 4);
    __builtin_amdgcn_global_load_async_to_lds_b128((gv4p)g, (lv4p)l, 0, 0);
  }
  asm volatile("s_wait_asynccnt 0" ::: "memory");
#else
  for (int c = tid; c < TOT_CH; c += BLOCK) {
    const int f = c / CH_PER_ROW;
    const int r = c - f * CH_PER_ROW;
    const float4* g =
        (const float4*)(tbase + (size_t)s_rowoff[f] + (size_t)((unsigned)r << 4));
    *(float4*)((char*)s_rows + ((size_t)c << 4)) = *g;
  }
#endif
  __syncthreads();

  // Phase 3: out[b,p,e] = table[x_i, j, e] * table[x_j, i, e], float4 at a
  // time. Output is streamed and never re-read -> non-temporal stores so the
  // 194MB of writes don't evict the table from L2.
  float* outb = out + (size_t)b * (NPAIR * EMB);
  for (int q = tid; q < OUT_F4; q += BLOCK) {
    const int p  = q >> 2;
    const int e  = (q & 3) << 2;
    const int ij = c_pairs.v[p];
    const int i  = ij >> 8;
    const int j  = ij & 0xFF;
    const v4f L = *(const v4f*)&s_rows[i * ROWF + j * EMB + e];
    const v4f R = *(const v4f*)&s_rows[j * ROWF + i * EMB + e];
    const v4f o = L * R;
    __builtin_nontemporal_store(o, (v4f*)(outb + ((size_t)q << 2)));
  }
}

extern "C" void kernel_launch(void* const* d_in, const int* in_sizes, int n_in,
                              void* d_out, int out_size, void* d_ws, size_t ws_size,
                              hipStream_t stream) {
  const int*   X     = (const int*)d_in[0];
  const float* table = (const float*)d_in[1];
  float*       out   = (float*)d_out;
  const int batch = in_sizes[0] / FIELDS;  // 4096
  ffm_interact_kernel<<<batch, BLOCK, 0, stream>>>(X, table, out);
  (void)n_in; (void)out_size; (void)d_ws; (void)ws_size;
}